// MatryoshkaSAE_61821759259158
// MI455X (gfx1250) — compile-verified
//
#include <hip/hip_runtime.h>
#include <hip/hip_bf16.h>

// Problem dims (fixed by reference)
#define BB      8
#define SS      2048
#define M_TOT   (BB * SS)   // 16384 rows
#define DMODEL  1024        // K of encoder GEMM, output dim of decoder
#define DLAT    4096        // N of encoder GEMM (latents)
#define TOPK    16

typedef __attribute__((ext_vector_type(2))) float v2f;
typedef __attribute__((ext_vector_type(4))) float v4f;
typedef __attribute__((ext_vector_type(8))) float v8f;
typedef __attribute__((ext_vector_type(4))) int   v4i;

typedef __attribute__((address_space(1))) v4i* gv4i_p;   // global b128
typedef __attribute__((address_space(3))) v4i* lv4i_p;   // LDS b128

// ---- CDNA5 async global->LDS copy (ASYNCcnt path), with safe fallback -----
#if __has_builtin(__builtin_amdgcn_global_load_async_to_lds_b128)
#define HAVE_ASYNC_LDS 1
#else
#define HAVE_ASYNC_LDS 0
#endif

__device__ __forceinline__ void async_copy16(const float* gp, float* lp) {
#if HAVE_ASYNC_LDS
  __builtin_amdgcn_global_load_async_to_lds_b128(
      (gv4i_p)(unsigned long long)gp, (lv4i_p)lp, 0, 0);
#else
  float4 v = *(const float4*)gp;
  lp[0] = v.x; lp[1] = v.y; lp[2] = v.z; lp[3] = v.w;
#endif
}

__device__ __forceinline__ void wait_async0() {
#if HAVE_ASYNC_LDS
#if __has_builtin(__builtin_amdgcn_s_wait_asynccnt)
  __builtin_amdgcn_s_wait_asynccnt(0);
#else
  asm volatile("s_wait_asynccnt 0x0" ::: "memory");
#endif
#endif
}

// ---------------------------------------------------------------------------
// Kernel 1: transpose W_dec [DMODEL][DLAT] -> W_decT [DLAT][DMODEL]
// ---------------------------------------------------------------------------
__global__ __launch_bounds__(256) void sae_transpose_wdec(
    const float* __restrict__ Wdec, float* __restrict__ WdecT) {
  __shared__ float t[32][33];
  int d0 = blockIdx.x * 32;            // over DMODEL
  int l0 = blockIdx.y * 32;            // over DLAT
  int lx = threadIdx.x & 31;
  int ly = threadIdx.x >> 5;           // 0..7
  for (int r = ly; r < 32; r += 8)
    t[r][lx] = Wdec[(size_t)(d0 + r) * DLAT + l0 + lx];
  __syncthreads();
  for (int r = ly; r < 32; r += 8)
    WdecT[(size_t)(l0 + r) * DMODEL + d0 + lx] = t[lx][r];
}

// ---------------------------------------------------------------------------
// Kernel 2: zero the dense latents output region (scatter background).
// Non-temporal: 268 MB of streaming writes must not evict x/W_enc from L2.
// ---------------------------------------------------------------------------
__global__ void sae_zero_f4(v4f* __restrict__ p, size_t n4) {
  size_t i = (size_t)blockIdx.x * blockDim.x + threadIdx.x;
  size_t stride = (size_t)gridDim.x * blockDim.x;
  v4f z = {0.f, 0.f, 0.f, 0.f};
  for (; i < n4; i += stride) __builtin_nontemporal_store(z, p + i);
}

// ---------------------------------------------------------------------------
// Kernel 3: fused encoder GEMM (fp32 WMMA 16x16x4) + per-row top-16 + relu
// + scatter into dense latents. Block = 256 threads = 8 wave32s, owns 32 rows.
// Double-buffered LDS tiles filled by async global->LDS copies overlap the
// WMMA compute phase; s_wait_asynccnt + workgroup barrier close each stage.
// ---------------------------------------------------------------------------
__global__ __launch_bounds__(256) void sae_enc_topk(
    const float* __restrict__ x,        // [M_TOT][DMODEL]
    const float* __restrict__ W_enc,    // [DLAT][DMODEL]
    const float* __restrict__ b_enc,    // [DLAT]
    const float* __restrict__ enc_bias, // [DLAT]
    float* __restrict__ latents_out,    // [M_TOT][DLAT] (pre-zeroed)
    float* __restrict__ topk_vals,      // [M_TOT][TOPK] (relu'd)
    int*   __restrict__ topk_idx) {     // [M_TOT][TOPK]
  constexpr int MT = 32;                // rows per block
  constexpr int NT = 64;                // latent-column chunk
  constexpr int KC = 64;                // K staging chunk
  constexpr int NKC = DMODEL / KC;      // 16
  constexpr int XST = KC + 4;           // stride % 64 == 4 -> conflict-free
  constexpr int WST = KC + 4;

  __shared__ __align__(16) float xs[2][MT * XST];   // 2 x  8.5 KB x chunk
  __shared__ __align__(16) float wsm[2][NT * WST];  // 2 x 17.0 KB W_enc chunk
  __shared__ float chunk[MT][NT + 1];               //  8.3 KB biased pre-acts
  __shared__ float tv[MT][TOPK];                    //  running top-k values
  __shared__ int   ti[MT][TOPK];                    //  running top-k indices

  const int tid  = threadIdx.x;
  const int m0   = blockIdx.x * MT;
  const int wave = tid >> 5;            // 0..7
  const int lane = tid & 31;
  const int h    = lane >> 4;           // half-wave select (K / row split)
  const int p    = lane & 15;           // M (for A) / N (for B,D) position
  const int mt   = wave >> 2;           // 0..1 : 16-row band
  const int nt   = wave & 3;            // 0..3 : 16-col band

  for (int f = tid; f < MT * TOPK; f += 256) {
    tv[f >> 4][f & 15] = -3.4e38f;
    ti[f >> 4][f & 15] = 0;
  }

  // Stage one (x, W_enc) K-chunk into LDS buffer `buf` (async if available).
  auto stage = [&](int buf, int k0, int n0) {
    // x chunk: 32 rows x 64 floats = 512 x b128, coalesced
    for (int f = tid; f < MT * (KC / 4); f += 256) {
      int row = f >> 4, c4 = f & 15;
      async_copy16(x + (size_t)(m0 + row) * DMODEL + k0 + c4 * 4,
                   &xs[buf][row * XST + c4 * 4]);
    }
    // W_enc chunk: 64 rows x 64 floats = 1024 x b128, coalesced
    for (int f = tid; f < NT * (KC / 4); f += 256) {
      int row = f >> 4, c4 = f & 15;
      async_copy16(W_enc + (size_t)(n0 + row) * DMODEL + k0 + c4 * 4,
                   &wsm[buf][row * WST + c4 * 4]);
    }
  };

  for (int n0 = 0; n0 < DLAT; n0 += NT) {
    v8f acc = {0.f, 0.f, 0.f, 0.f, 0.f, 0.f, 0.f, 0.f};

    stage(0, 0, n0);          // prologue fill
    wait_async0();
    __syncthreads();

    for (int kc = 0; kc < NKC; ++kc) {
      const int cur = kc & 1;
      if (kc + 1 < NKC) stage(cur ^ 1, (kc + 1) * KC, n0);  // overlap copy

      const float* arow = &xs[cur][(mt * 16 + p) * XST];  // A: lane p = row M
      const float* brow = &wsm[cur][(nt * 16 + p) * WST]; // B: lane p = col N
      #pragma unroll
      for (int kk = 0; kk < KC; kk += 4) {
        v2f a = { arow[kk + 2 * h], arow[kk + 2 * h + 1] };  // K = 2h, 2h+1
        v2f b = { brow[kk + 2 * h], brow[kk + 2 * h + 1] };
        acc = __builtin_amdgcn_wmma_f32_16x16x4_f32(
            false, a, false, b, (short)0, acc, false, false);
      }

      wait_async0();          // next buffer's LDS writes landed (this wave)
      __syncthreads();        // ... and everyone else's
    }

    // Bias + scatter D fragments (lane = 16h+p holds col p, rows r+8h)
    const int col = n0 + nt * 16 + p;
    const float bv = b_enc[col] + enc_bias[col];
    #pragma unroll
    for (int r = 0; r < 8; ++r)
      chunk[mt * 16 + r + 8 * h][nt * 16 + p] = acc[r] + bv;
    __syncthreads();

    // Per-row top-16 maintenance: thread r owns row r (sorted descending)
    if (tid < MT) {
      float cmin = tv[tid][TOPK - 1];
      for (int c = 0; c < NT; ++c) {
        float v = chunk[tid][c];
        if (v > cmin) {
          int pos = TOPK - 1;
          while (pos > 0 && tv[tid][pos - 1] < v) {
            tv[tid][pos] = tv[tid][pos - 1];
            ti[tid][pos] = ti[tid][pos - 1];
            --pos;
          }
          tv[tid][pos] = v;
          ti[tid][pos] = n0 + c;
          cmin = tv[tid][TOPK - 1];
        }
      }
    }
  }
  __syncthreads();

  // Emit: relu'd top-k -> dense latents scatter + compact (val,idx) for decode
  if (tid < MT) {
    size_t m = (size_t)(m0 + tid);
    for (int k = 0; k < TOPK; ++k) {
      float v = tv[tid][k];
      float rv = v > 0.f ? v : 0.f;
      int l = ti[tid][k];
      __builtin_nontemporal_store(rv, &latents_out[m * DLAT + (size_t)l]);
      topk_vals[m * TOPK + k] = rv;
      topk_idx[m * TOPK + k]  = l;
    }
  }
}

// ---------------------------------------------------------------------------
// Kernel 4: sparse decode  recon[m,:] = sum_k val_k * W_decT[idx_k, :]
// One block per row; W_decT rows are contiguous & L2-resident.
// ---------------------------------------------------------------------------
__global__ __launch_bounds__(256) void sae_decode(
    const float* __restrict__ topk_vals, const int* __restrict__ topk_idx,
    const float* __restrict__ WdecT, float* __restrict__ recon) {
  __shared__ float sv[TOPK];
  __shared__ int   si[TOPK];
  const size_t m = blockIdx.x;
  const int tid = threadIdx.x;
  if (tid < TOPK) {
    sv[tid] = topk_vals[m * TOPK + tid];
    si[tid] = topk_idx[m * TOPK + tid];
  }
  __syncthreads();
  v4f acc = {0.f, 0.f, 0.f, 0.f};
  #pragma unroll
  for (int k = 0; k < TOPK; ++k) {
    float w = sv[k];
    v4f c = ((const v4f*)(WdecT + (size_t)si[k] * DMODEL))[tid];
    acc += w * c;
  }
  __builtin_nontemporal_store(acc, ((v4f*)(recon + m * DMODEL)) + tid);
}

// ---------------------------------------------------------------------------
extern "C" void kernel_launch(void* const* d_in, const int* in_sizes, int n_in,
                              void* d_out, int out_size, void* d_ws, size_t ws_size,
                              hipStream_t stream) {
  const float* x        = (const float*)d_in[0];
  const float* W_enc    = (const float*)d_in[1];
  const float* b_enc    = (const float*)d_in[2];
  const float* enc_bias = (const float*)d_in[3];
  const float* W_dec    = (const float*)d_in[4];

  float* recon   = (float*)d_out;                                 // [M_TOT][DMODEL]
  float* latents = (float*)d_out + (size_t)M_TOT * DMODEL;        // [M_TOT][DLAT]

  float* WdecT = (float*)d_ws;                                    // 16.8 MB
  float* tkv   = WdecT + (size_t)DLAT * DMODEL;                   //  1 MB
  int*   tki   = (int*)(tkv + (size_t)M_TOT * TOPK);              //  1 MB

  // 1) W_dec transpose for contiguous sparse-decode rows
  dim3 tgrid(DMODEL / 32, DLAT / 32);
  sae_transpose_wdec<<<tgrid, 256, 0, stream>>>(W_dec, WdecT);

  // 2) zero dense latents region of d_out (scatter background, NT stores)
  size_t n4 = (size_t)M_TOT * DLAT / 4;
  sae_zero_f4<<<4096, 256, 0, stream>>>((v4f*)latents, n4);

  // 3) fused fp32-WMMA encoder GEMM + top-k + relu + scatter
  sae_enc_topk<<<M_TOT / 32, 256, 0, stream>>>(
      x, W_enc, b_enc, enc_bias, latents, tkv, tki);

  // 4) sparse decode
  sae_decode<<<M_TOT, 256, 0, stream>>>(tkv, tki, WdecT, recon);
}